// FusedTensorProductOp4_55808805044383
// MI455X (gfx1250) — compile-verified
//
#include <hip/hip_runtime.h>
#include <hip/hip_bf16.h>

// FusedTensorProductOp4 for MI455X (gfx1250, wave32).
//
// out[b, 128k+w] = sum_u A_k[b,u] * X2_k[b,u,w]
//   A_k[b,u]     = sum_{(i,j):(i+j)%2==k} c_{8i+j} * in0[b,128i+u]*in1[b,128j+u]
//
// Memory-bound (~545 MB traffic, ~0.4 GFLOP -> ~0.7 flop/byte; HBM floor at
// 23.3 TB/s is ~23 us). One workgroup per (batch, parity) streams its 64 KB
// X2 matrix exactly once with non-temporal loads. The vec-mat MACs are folded
// into V_WMMA_F32_16X16X4_F32 by replicating the weight vector across all 16
// A-rows: redundant matrix flops are free at this intensity, fp32 precision
// is preserved, and the accumulator lives in the WMMA pipe for all 32
// K-chunks. grid = (4096, 2) x 128 threads = 8192 concurrent 64 KB streams.

typedef __attribute__((ext_vector_type(2))) float v2f;
typedef __attribute__((ext_vector_type(8))) float v8f;

#define U_DIM 128
#define W_DIM 128
#define IN_COLS 1024        // U * NSEG_IN
#define IN2_SEG 16384       // U * W
#define OUT_STRIDE 256
#define PF_DIST (16 * W_DIM)   // prefetch 4 u-chunks (16 rows) ahead

__global__ __launch_bounds__(128)
void ftp4_kernel(const float* __restrict__ in0,
                 const float* __restrict__ in1,
                 const float* __restrict__ in2,
                 float* __restrict__ out) {
    const int b = blockIdx.x;
    const int k = blockIdx.y;             // parity: 0 or 1
    const int t = threadIdx.x;            // 0..127, also the u index in phase 1

    __shared__ float Aw[U_DIM];           // weight vector for this parity

    // ------ Phase 1: bilinear weights for parity k (coeffs folded in) -----
    {
        float x0[8], x1[8];
        const float* p0 = in0 + (size_t)b * IN_COLS + t;
        const float* p1 = in1 + (size_t)b * IN_COLS + t;
#pragma unroll
        for (int i = 0; i < 8; ++i) {
            x0[i] = p0[i * U_DIM];
            x1[i] = p1[i * U_DIM];
        }
        float a = 0.f;
#pragma unroll
        for (int i = 0; i < 8; ++i) {
#pragma unroll
            for (int j = 0; j < 8; ++j) {
                if (((i + j) & 1) != 0 == (k == 0)) continue;  // keep parity k
                const int p = i * 8 + j;
                const float c = ((p & 1) ? -1.f : 1.f) * (0.1f + 0.01f * (float)p);
                a += c * x0[i] * x1[j];
            }
        }
        Aw[t] = a;
    }
    __syncthreads();

    // ------ Phase 2: WMMA-accumulated vec-mat stream -----------------------
    // Wave wv owns output columns [32wv, 32wv+32) as two 16-wide WMMA tiles.
    // A(16x4 f32) layout: lane<16 -> K=0,1 ; lane>=16 -> K=2,3  (2 VGPRs)
    // B(4x16  f32) layout: same K split; N = lane & 15.
    const int lane  = t & 31;
    const int wv    = t >> 5;             // wave id 0..3
    const int wbase = wv * 32;
    const int half  = lane >> 4;          // 0: K=0,1   1: K=2,3
    const int n     = lane & 15;

    const float* X2 = in2 + (size_t)b * (2 * IN2_SEG) + (size_t)k * IN2_SEG;

    v8f acc1 = {};                        // tile at w = wbase
    v8f acc2 = {};                        // tile at w = wbase + 16

    // per-lane pointers into X2: row = half*2, col = wbase + n
    const float* q1 = X2 + (size_t)(half * 2) * W_DIM + wbase + n;
    const float* q2 = q1 + 16;

#pragma unroll 4
    for (int ub = 0; ub < U_DIM; ub += 4) {
        // replicated A: lanes<16 get (A[ub],A[ub+1]), lanes>=16 (A[ub+2],A[ub+3])
        v2f a;
        a.x = Aw[ub + half * 2];
        a.y = Aw[ub + half * 2 + 1];

        // single-use stream: non-temporal so 512 MB doesn't thrash L2
        v2f b1, b2;
        b1.x = __builtin_nontemporal_load(q1);
        b1.y = __builtin_nontemporal_load(q1 + W_DIM);
        b2.x = __builtin_nontemporal_load(q2);
        b2.y = __builtin_nontemporal_load(q2 + W_DIM);

        // stage all four upcoming rows (r, r+2 via lane halves; r+1, r+3
        // via the +W_DIM variant) into GL2 ~4 chunks ahead
        __builtin_prefetch(q1 + PF_DIST, 0, 0);
        __builtin_prefetch(q1 + PF_DIST + W_DIM, 0, 0);

        acc1 = __builtin_amdgcn_wmma_f32_16x16x4_f32(
                   false, a, false, b1, (short)0, acc1, false, false);
        acc2 = __builtin_amdgcn_wmma_f32_16x16x4_f32(
                   false, a, false, b2, (short)0, acc2, false, false);

        q1 += 4 * W_DIM;
        q2 += 4 * W_DIM;
    }

    // All D rows are identical; element 0 holds the result for
    // w = tile_base + n. Lanes 0-15 emit tile1, 16-31 tile2, so the wave
    // writes the contiguous 32-float slice [wbase, wbase+32).
    const float val = (half == 0) ? acc1[0] : acc2[0];
    out[(size_t)b * OUT_STRIDE + (size_t)k * W_DIM + wbase + lane] = val;
}

extern "C" void kernel_launch(void* const* d_in, const int* in_sizes, int n_in,
                              void* d_out, int out_size, void* d_ws, size_t ws_size,
                              hipStream_t stream) {
    const float* in0 = (const float*)d_in[0];
    const float* in1 = (const float*)d_in[1];
    const float* in2 = (const float*)d_in[2];
    float* out = (float*)d_out;

    const int Z = in_sizes[0] / IN_COLS;   // 4096
    ftp4_kernel<<<dim3(Z, 2), dim3(128), 0, stream>>>(in0, in1, in2, out);
}